// FLAME_67654324847409
// MI455X (gfx1250) — compile-verified
//
#include <hip/hip_runtime.h>
#include <hip/hip_bf16.h>
#include <math.h>

typedef float v2f __attribute__((ext_vector_type(2)));
typedef float v8f __attribute__((ext_vector_type(8)));

// ---------------- problem constants ----------------
#define BSZ   1024
#define NV    5023
#define NF    9976
#define NJ    5
#define NCOLS (NV * 3)          // 15069
#define NPAD  15104             // NCOLS padded to 64 (for 64-wide wave tiles)
#define KPAD  192               // 150 (shape) + 36 (pose) padded to 192
#define WT_M  64                // wave tile M
#define WT_N  64                // wave tile N
#define NWT_M (BSZ / WT_M)      // 16
#define NWT_N (NPAD / WT_N)     // 236

// ---------------- workspace layout (floats) ----------------
#define WS_APAD   0
#define WS_ROT    (WS_APAD + BSZ * KPAD)
#define WS_AMATS  (WS_ROT + BSZ * 45)
#define WS_YIDX   (WS_AMATS + BSZ * 80)
#define WS_JS     (WS_YIDX + BSZ)
#define WS_BPAD   (WS_JS + 4096)
#define WS_VPOSED (WS_BPAD + KPAD * NPAD)

__device__ __forceinline__ void rodrigues(const float* v, float* R) {
    // SMPL convention: angle = ||v + 1e-8||, dir = v / angle
    float px = v[0] + 1e-8f, py = v[1] + 1e-8f, pz = v[2] + 1e-8f;
    float ang = sqrtf(px * px + py * py + pz * pz);
    float rx = v[0] / ang, ry = v[1] / ang, rz = v[2] / ang;
    float c = cosf(ang), s = sinf(ang), t = 1.0f - c;
    // R = c*I + s*K + (1-c)*r r^T
    R[0] = c + t * rx * rx;       R[1] = -s * rz + t * rx * ry; R[2] =  s * ry + t * rx * rz;
    R[3] =  s * rz + t * ry * rx; R[4] = c + t * ry * ry;       R[5] = -s * rx + t * ry * rz;
    R[6] = -s * ry + t * rz * rx; R[7] =  s * rx + t * rz * ry; R[8] = c + t * rz * rz;
}

// ---------------- kernel 1: per-batch pose prep ----------------
__global__ void pose_kernel(const float* __restrict__ shape_p,
                            const float* __restrict__ expr_p,
                            const float* __restrict__ pose_p,
                            const float* __restrict__ eye_p,
                            float* __restrict__ ws) {
    int b = blockIdx.x * blockDim.x + threadIdx.x;
    if (b >= BSZ) return;

    float fp[15];
    fp[0] = pose_p[b * 6 + 0]; fp[1] = pose_p[b * 6 + 1]; fp[2] = pose_p[b * 6 + 2];
    fp[3] = 0.0f; fp[4] = 0.0f; fp[5] = 0.0f;                       // neck pose = 0
    fp[6] = pose_p[b * 6 + 3]; fp[7] = pose_p[b * 6 + 4]; fp[8] = pose_p[b * 6 + 5];
    for (int i = 0; i < 6; ++i) fp[9 + i] = eye_p[b * 6 + i];

    float R[NJ][9];
    for (int j = 0; j < NJ; ++j) rodrigues(&fp[j * 3], R[j]);

    float* rot = ws + WS_ROT + (size_t)b * 45;
    for (int j = 0; j < NJ; ++j)
        for (int e = 0; e < 9; ++e) rot[j * 9 + e] = R[j][e];

    // padded A row: [shape(100) | expr(50) | pose_feature(36) | zeros(6)]
    float* arow = ws + WS_APAD + (size_t)b * KPAD;
    for (int l = 0; l < 100; ++l) arow[l] = shape_p[b * 100 + l];
    for (int l = 0; l < 50; ++l) arow[100 + l] = expr_p[b * 50 + l];
    for (int j = 1; j < NJ; ++j)
        for (int e = 0; e < 9; ++e) {
            float iv = (e == 0 || e == 4 || e == 8) ? 1.0f : 0.0f;
            arow[150 + (j - 1) * 9 + e] = R[j][e] - iv;
        }
    for (int l = 186; l < KPAD; ++l) arow[l] = 0.0f;

    // dynamic landmark index: rel = R[0] @ R[1]   (NECK_CHAIN = [1, 0])
    float rel[9];
    for (int r = 0; r < 3; ++r)
        for (int cI = 0; cI < 3; ++cI) {
            float s = 0.0f;
            for (int k = 0; k < 3; ++k) s += R[0][r * 3 + k] * R[1][k * 3 + cI];
            rel[r * 3 + cI] = s;
        }
    float sy = sqrtf(rel[0] * rel[0] + rel[3] * rel[3]);
    float y_deg = atan2f(-rel[6], sy) * 57.29577951308232f;
    y_deg = fminf(y_deg, 39.0f);
    int y = (int)rintf(y_deg);               // round-to-nearest-even
    if (y < 0) y = (y < -39) ? 78 : 39 - y;
    ((int*)(ws + WS_YIDX))[b] = y;
}

// ---------------- kernel 2: build padded B (K x Npad, row-major) ----------------
__global__ void bpad_kernel(const float* __restrict__ shapedirs,
                            const float* __restrict__ posedirs,
                            float* __restrict__ ws) {
    size_t idx = (size_t)blockIdx.x * blockDim.x + threadIdx.x;
    if (idx >= (size_t)KPAD * NPAD) return;
    int k = (int)(idx / NPAD);
    int n = (int)(idx % NPAD);
    float val = 0.0f;
    if (n < NCOLS) {
        if (k < 150)        val = shapedirs[(size_t)n * 150 + k];      // (V,3,150) == (N,K)
        else if (k < 186)   val = posedirs[(size_t)(k - 150) * NCOLS + n];
    }
    ws[WS_BPAD + idx] = val;
}

// ---------------- kernel 3: J_regressor @ [shapedirs | v_template] ----------------
__global__ void js_kernel(const float* __restrict__ J_reg,
                          const float* __restrict__ shapedirs,
                          const float* __restrict__ v_template,
                          float* __restrict__ ws) {
    int idx = blockIdx.x * blockDim.x + threadIdx.x;
    if (idx >= 2265) return;
    float sum = 0.0f;
    if (idx < 2250) {                 // JS[j][c][l]
        int j = idx / 450, rem = idx % 450, cI = rem / 150, l = rem % 150;
        const float* Jr = J_reg + (size_t)j * NV;
        for (int v = 0; v < NV; ++v)
            sum += Jr[v] * shapedirs[((size_t)v * 3 + cI) * 150 + l];
    } else {                          // JT[j][c]
        int t = idx - 2250, j = t / 3, cI = t % 3;
        const float* Jr = J_reg + (size_t)j * NV;
        for (int v = 0; v < NV; ++v)
            sum += Jr[v] * v_template[(size_t)v * 3 + cI];
    }
    ws[WS_JS + idx] = sum;
}

// ---------------- kernel 4: FP32 WMMA GEMM, 64x64 wave tile ----------------
// v_posed(B x Npad) = Apad(B x K) @ Bpad(K x Npad) + v_template
__global__ void __launch_bounds__(256)
wmma_gemm_kernel(const float* __restrict__ ws_in,
                 const float* __restrict__ v_template,
                 float* __restrict__ ws_out) {
    const float* Apad = ws_in + WS_APAD;
    const float* Bpad = ws_in + WS_BPAD;
    float* vpos = ws_out + WS_VPOSED;

    int wtile = blockIdx.x * 8 + (threadIdx.x >> 5);   // grid sized exactly: no divergence
    int mt = wtile / NWT_N;
    int nt = wtile % NWT_N;
    int lane = threadIdx.x & 31;
    int m0 = mt * WT_M, n0 = nt * WT_N;
    int row = lane & 15;
    int khalf = (lane >> 4) << 1;                      // 0 or 2

    v8f acc[4][4];
    #pragma unroll
    for (int mi = 0; mi < 4; ++mi)
        #pragma unroll
        for (int ni = 0; ni < 4; ++ni)
            acc[mi][ni] = (v8f){};

    const float* aBase = Apad + (size_t)(m0 + row) * KPAD + khalf;
    const float* bBase = Bpad + (size_t)khalf * NPAD + n0 + row;

    #pragma unroll 2
    for (int k = 0; k < KPAD; k += 4) {
        v2f a[4], bf[4];
        #pragma unroll
        for (int mi = 0; mi < 4; ++mi) {
            const float* ap = aBase + (size_t)mi * 16 * KPAD + k;
            a[mi].x = ap[0];
            a[mi].y = ap[1];
        }
        #pragma unroll
        for (int ni = 0; ni < 4; ++ni) {
            const float* bp = bBase + (size_t)k * NPAD + ni * 16;
            bf[ni].x = bp[0];
            bf[ni].y = bp[NPAD];
        }
        #pragma unroll
        for (int mi = 0; mi < 4; ++mi)
            #pragma unroll
            for (int ni = 0; ni < 4; ++ni)
                acc[mi][ni] = __builtin_amdgcn_wmma_f32_16x16x4_f32(
                    false, a[mi], false, bf[ni], (short)0, acc[mi][ni], false, false);
    }

    // epilogue: C layout per 16x16 subtile: VGPR r -> M = r (+8 for lanes 16-31), N = lane&15
    int half8 = (lane >> 4) << 3;
    #pragma unroll
    for (int ni = 0; ni < 4; ++ni) {
        int ncol = n0 + ni * 16 + row;
        float vt = (ncol < NCOLS) ? v_template[ncol] : 0.0f;
        #pragma unroll
        for (int mi = 0; mi < 4; ++mi) {
            int mbase = m0 + mi * 16 + half8;
            #pragma unroll
            for (int r = 0; r < 8; ++r)
                vpos[(size_t)(mbase + r) * NPAD + ncol] = acc[mi][ni][r] + vt;
        }
    }
}

// ---------------- kernel 5: joints + kinematic chain -> skinning mats ----------------
__global__ void transforms_kernel(float* __restrict__ ws) {
    int b = blockIdx.x * blockDim.x + threadIdx.x;
    if (b >= BSZ) return;
    const int parents[NJ] = {-1, 0, 1, 1, 1};

    const float* arow = ws + WS_APAD + (size_t)b * KPAD;   // betas in [0,150)
    const float* JS = ws + WS_JS;
    const float* JT = ws + WS_JS + 2250;
    const float* rot = ws + WS_ROT + (size_t)b * 45;

    float joints[NJ][3];
    for (int j = 0; j < NJ; ++j)
        for (int cI = 0; cI < 3; ++cI) {
            float s = JT[j * 3 + cI];
            const float* row = JS + (size_t)(j * 3 + cI) * 150;
            for (int l = 0; l < 150; ++l) s += row[l] * arow[l];
            joints[j][cI] = s;
        }

    float G[NJ][16];
    for (int j = 0; j < NJ; ++j) {
        float rel[3];
        if (j == 0) { rel[0] = joints[0][0]; rel[1] = joints[0][1]; rel[2] = joints[0][2]; }
        else {
            int p = parents[j];
            rel[0] = joints[j][0] - joints[p][0];
            rel[1] = joints[j][1] - joints[p][1];
            rel[2] = joints[j][2] - joints[p][2];
        }
        float tm[16];
        for (int r = 0; r < 3; ++r) {
            tm[r * 4 + 0] = rot[j * 9 + r * 3 + 0];
            tm[r * 4 + 1] = rot[j * 9 + r * 3 + 1];
            tm[r * 4 + 2] = rot[j * 9 + r * 3 + 2];
            tm[r * 4 + 3] = rel[r];
        }
        tm[12] = 0.0f; tm[13] = 0.0f; tm[14] = 0.0f; tm[15] = 1.0f;

        if (j == 0) {
            for (int e = 0; e < 16; ++e) G[0][e] = tm[e];
        } else {
            const float* Gp = G[parents[j]];
            for (int r = 0; r < 4; ++r)
                for (int cI = 0; cI < 4; ++cI) {
                    float s = 0.0f;
                    for (int k = 0; k < 4; ++k) s += Gp[r * 4 + k] * tm[k * 4 + cI];
                    G[j][r * 4 + cI] = s;
                }
        }
    }

    float* Ab = ws + WS_AMATS + (size_t)b * 80;
    for (int j = 0; j < NJ; ++j) {
        float jx = joints[j][0], jy = joints[j][1], jz = joints[j][2];
        for (int r = 0; r < 4; ++r) {
            float ib = G[j][r * 4 + 0] * jx + G[j][r * 4 + 1] * jy + G[j][r * 4 + 2] * jz;
            Ab[j * 16 + r * 4 + 0] = G[j][r * 4 + 0];
            Ab[j * 16 + r * 4 + 1] = G[j][r * 4 + 1];
            Ab[j * 16 + r * 4 + 2] = G[j][r * 4 + 2];
            Ab[j * 16 + r * 4 + 3] = G[j][r * 4 + 3] - ib;
        }
    }
}

// ---------------- kernel 6: LBS skinning (streaming) ----------------
__global__ void verts_kernel(const float* __restrict__ ws,
                             const float* __restrict__ lbs_w,
                             float* __restrict__ out) {
    size_t idx = (size_t)blockIdx.x * blockDim.x + threadIdx.x;
    if (idx >= (size_t)BSZ * NV) return;
    int b = (int)(idx / NV);
    int v = (int)(idx % NV);

    const float* Ab = ws + WS_AMATS + (size_t)b * 80;
    float w[NJ];
    #pragma unroll
    for (int j = 0; j < NJ; ++j) w[j] = lbs_w[(size_t)v * NJ + j];

    float T[12];
    #pragma unroll
    for (int e = 0; e < 12; ++e) {
        float s = 0.0f;
        #pragma unroll
        for (int j = 0; j < NJ; ++j) s += w[j] * Ab[j * 16 + e];
        T[e] = s;
    }

    const float* p = ws + WS_VPOSED + (size_t)b * NPAD + (size_t)v * 3;
    float px = p[0], py = p[1], pz = p[2];
    float* o = out + ((size_t)b * NV + v) * 3;
    o[0] = T[0] * px + T[1] * py + T[2]  * pz + T[3];
    o[1] = T[4] * px + T[5] * py + T[6]  * pz + T[7];
    o[2] = T[8] * px + T[9] * py + T[10] * pz + T[11];
}

// ---------------- kernel 7: landmarks (dyn + static 2d, full 3d) ----------------
__global__ void lmk_kernel(const float* __restrict__ ws,
                           const int* __restrict__ faces,
                           const int* __restrict__ lmk_faces_idx,
                           const float* __restrict__ lmk_bary,
                           const int* __restrict__ dyn_faces,
                           const float* __restrict__ dyn_bary,
                           const int* __restrict__ full_faces,
                           const float* __restrict__ full_bary,
                           float* __restrict__ out) {
    int idx = blockIdx.x * blockDim.x + threadIdx.x;
    const int total = BSZ * 68 * 2;
    if (idx >= total) return;
    int which = idx / (BSZ * 68);
    int rem = idx % (BSZ * 68);
    int b = rem / 68;
    int l = rem % 68;

    int f;
    float bw0, bw1, bw2;
    size_t ooff;
    if (which == 0) {
        if (l < 17) {
            int y = ((const int*)(ws + WS_YIDX))[b];
            int e = y * 17 + l;
            f = dyn_faces[e];
            bw0 = dyn_bary[e * 3 + 0]; bw1 = dyn_bary[e * 3 + 1]; bw2 = dyn_bary[e * 3 + 2];
        } else {
            int e = l - 17;
            f = lmk_faces_idx[e];
            bw0 = lmk_bary[e * 3 + 0]; bw1 = lmk_bary[e * 3 + 1]; bw2 = lmk_bary[e * 3 + 2];
        }
        ooff = (size_t)BSZ * NV * 3 + ((size_t)b * 68 + l) * 3;
    } else {
        f = full_faces[l];
        bw0 = full_bary[l * 3 + 0]; bw1 = full_bary[l * 3 + 1]; bw2 = full_bary[l * 3 + 2];
        ooff = (size_t)BSZ * NV * 3 + (size_t)BSZ * 68 * 3 + ((size_t)b * 68 + l) * 3;
    }

    const float* verts = out;  // vertices already written at base of d_out
    int v0 = faces[f * 3 + 0], v1 = faces[f * 3 + 1], v2 = faces[f * 3 + 2];
    const float* p0 = verts + ((size_t)b * NV + v0) * 3;
    const float* p1 = verts + ((size_t)b * NV + v1) * 3;
    const float* p2 = verts + ((size_t)b * NV + v2) * 3;
    out[ooff + 0] = bw0 * p0[0] + bw1 * p1[0] + bw2 * p2[0];
    out[ooff + 1] = bw0 * p0[1] + bw1 * p1[1] + bw2 * p2[1];
    out[ooff + 2] = bw0 * p0[2] + bw1 * p1[2] + bw2 * p2[2];
}

// ---------------- launcher ----------------
extern "C" void kernel_launch(void* const* d_in, const int* in_sizes, int n_in,
                              void* d_out, int out_size, void* d_ws, size_t ws_size,
                              hipStream_t stream) {
    const float* shape_p    = (const float*)d_in[0];
    const float* expr_p     = (const float*)d_in[1];
    const float* pose_p     = (const float*)d_in[2];
    const float* eye_p      = (const float*)d_in[3];
    const float* v_template = (const float*)d_in[4];
    const float* shapedirs  = (const float*)d_in[5];
    const float* posedirs   = (const float*)d_in[6];
    const float* J_reg      = (const float*)d_in[7];
    const float* lbs_w      = (const float*)d_in[8];
    const int*   faces      = (const int*)d_in[9];
    const int*   lmk_f      = (const int*)d_in[10];
    const float* lmk_b      = (const float*)d_in[11];
    const int*   dyn_f      = (const int*)d_in[12];
    const float* dyn_b      = (const float*)d_in[13];
    const int*   full_f     = (const int*)d_in[14];
    const float* full_b     = (const float*)d_in[15];

    float* ws = (float*)d_ws;
    float* out = (float*)d_out;

    // 1. pose / Rodrigues / A-matrix rows / dynamic landmark index
    pose_kernel<<<(BSZ + 255) / 256, 256, 0, stream>>>(shape_p, expr_p, pose_p, eye_p, ws);

    // 2. materialize padded B (K x Npad)
    {
        size_t n = (size_t)KPAD * NPAD;
        bpad_kernel<<<(unsigned)((n + 255) / 256), 256, 0, stream>>>(shapedirs, posedirs, ws);
    }

    // 3. J_regressor contractions for joints
    js_kernel<<<(2265 + 255) / 256, 256, 0, stream>>>(J_reg, shapedirs, v_template, ws);

    // 4. WMMA GEMM (64x64 wave tiles): v_posed = [betas|pose_feat] @ B + v_template
    {
        int total_wtiles = NWT_M * NWT_N;         // 16 * 236 = 3776, divisible by 8
        wmma_gemm_kernel<<<total_wtiles / 8, 256, 0, stream>>>(ws, v_template, ws);
    }

    // 5. joints -> kinematic chain -> skinning matrices
    transforms_kernel<<<(BSZ + 255) / 256, 256, 0, stream>>>(ws);

    // 6. LBS skinning -> vertices
    {
        size_t n = (size_t)BSZ * NV;
        verts_kernel<<<(unsigned)((n + 255) / 256), 256, 0, stream>>>(ws, lbs_w, out);
    }

    // 7. landmarks
    lmk_kernel<<<(BSZ * 68 * 2 + 255) / 256, 256, 0, stream>>>(
        ws, faces, lmk_f, lmk_b, dyn_f, dyn_b, full_f, full_b, out);
}